// AttentionCrossChannel_14336600834767
// MI455X (gfx1250) — compile-verified
//
#include <hip/hip_runtime.h>
#include <math.h>

typedef __attribute__((ext_vector_type(2))) float v2f;
typedef __attribute__((ext_vector_type(8))) float v8f;
typedef __attribute__((ext_vector_type(4))) unsigned u32x4;
typedef __attribute__((ext_vector_type(8))) int i32x8;
typedef __attribute__((ext_vector_type(4))) int i32x4;

#define PS 68       // kernel-1 xbuf pixel-major stride: 4*r mod 64 -> b64 conflict-free
#define XS 340      // tbuf row stride (floats); 8*XS%64=32 -> conflict-free C stores
#define QS 257      // qc/kc row stride
#define VS 272      // kernel-3 v tile stride (= 256 + 16 TDM pad dwords)
#define NT1 21      // ceil(324/16) n-tiles over halo pixels

__device__ __forceinline__ v8f wmma_f32_4(v2f a, v2f b, v8f c) {
  // V_WMMA_F32_16X16X4_F32 : D = A(16x4 f32) * B(4x16 f32) + C(16x16 f32)
  return __builtin_amdgcn_wmma_f32_16x16x4_f32(false, a, false, b, (short)0, c,
                                               false, false);
}

// TDM: async-load a 64-row x 256-dword tile (row stride 65536 elems) into LDS,
// inserting 16 pad dwords after every 256 dwords (-> LDS row stride VS=272).
__device__ __forceinline__ void tdm_load_tile(const float* gsrc, unsigned lds_byte) {
  unsigned long long ga = (unsigned long long)(const void*)gsrc;
  u32x4 g0;
  g0[0] = 1u;                                              // count=1, user mode
  g0[1] = lds_byte;                                        // lds_addr
  g0[2] = (unsigned)(ga & 0xFFFFFFFFull);                  // global_addr[31:0]
  g0[3] = (unsigned)((ga >> 32) & 0x1FFFFFFull) | (2u << 30);  // ga[56:32], type=2
  i32x8 g1;
  g1[0] = (int)((2u << 16)      // data_size = 4B
              | (1u << 20)      // pad_enable
              | (7u << 22)      // pad_interval: 256 dwords
              | (15u << 25));   // pad_amount: 16 dwords
  g1[1] = 0;                    // tensor_dim0 = 65536 -> low16 = 0
  g1[2] = (int)(1u | (64u << 16));   // tensor_dim0 hi16=1 ; tensor_dim1 = 64
  g1[3] = (int)(256u << 16);         // tile_dim0 = 256
  g1[4] = 64;                        // tile_dim1 = 64, tile_dim2 = 0
  g1[5] = 65536;                     // tensor_dim0_stride
  g1[6] = 0;
  g1[7] = 0;
  i32x4 z4 = {0, 0, 0, 0};
  i32x8 z8 = {0, 0, 0, 0, 0, 0, 0, 0};
  __builtin_amdgcn_tensor_load_to_lds(g0, g1, z4, z4, z8, 0);
}

// ---------------------------------------------------------------------------
// Kernel 1: fused conv1x1(qkv) + dw3x3 for BOTH streams on a 16x16 pixel tile.
//  - writes v1/v2 to workspace
//  - accumulates per-head 8x8 Grams G1=q1.k2, G2=q2.k1 and raw L2 norms
//    of q1,k1,q2,k2 via global f32 atomics (q/k never touch HBM)
// ---------------------------------------------------------------------------
__global__ __launch_bounds__(256) void k_qkv(
    const float* __restrict__ xir, const float* __restrict__ xvi,
    const float* __restrict__ wqkv, const float* __restrict__ wdw,
    float* __restrict__ vout,   // [2][4][64][65536]
    float* __restrict__ G,      // [2][4][8][8][8]
    float* __restrict__ norms)  // [4][4][64]  (nq1,nk1,nq2,nk2)
{
  extern __shared__ float lds[];
  float* xbuf = lds;               // 336 x PS  (pixel-major halo: [hp][ic])
  float* tbuf = xbuf + 336 * PS;   // 64 x XS   (conv1x1 result, channel-major)
  float* qc   = tbuf + 64 * XS;    // 64 x QS   (q1 center tile)
  float* kc   = qc   + 64 * QS;    // 64 x QS   (k1 center tile)

  const int tid  = threadIdx.x;
  const int wg   = blockIdx.x;
  const int b    = wg >> 8;
  const int tile = wg & 255;
  const int ty = (tile >> 4) << 4, tx = (tile & 15) << 4;
  const int wave = tid >> 5, lane = tid & 31;
  const int m16 = lane & 15, kh = lane >> 4;

  for (int s = 0; s < 2; ++s) {
    const float* __restrict__ x = s ? xvi : xir;

    // ---- stage 18x18 halo of all 64 input channels (zero OOB == SAME pad) --
    for (int idx = tid; idx < 64 * 324; idx += 256) {
      int c = idx / 324, hp = idx - c * 324;
      int hy = hp / 18, hx = hp - hy * 18;
      int gy = ty + hy - 1, gx = tx + hx - 1;
      float val = 0.f;
      if ((unsigned)gy < 256u && (unsigned)gx < 256u)
        val = x[((size_t)(b * 64 + c) << 16) + (gy << 8) + gx];
      xbuf[hp * PS + c] = val;
    }
    __syncthreads();

    for (int g = 0; g < 3; ++g) {  // 0=q 1=k 2=v
      // ---- conv1x1 as WMMA GEMM: t[64 x 324] = W[g] (64x64) @ x ----------
      for (int t8 = wave; t8 < 4 * NT1; t8 += 8) {  // wave-uniform: EXEC full
        int mt = t8 / NT1, nt = t8 - mt * NT1;
        int ocb = g * 64 + mt * 16;
        int nb = nt * 16;
        const float* warow = &wqkv[(ocb + m16) * 64];   // A row for this lane
        const float* xrow  = &xbuf[(nb + m16) * PS];    // B col for this lane
        v8f acc = {};
        for (int k0 = 0; k0 < 64; k0 += 4) {
          int kk = k0 + kh * 2;
          v2f a  = *(const v2f*)(warow + kk);           // global_load_b64
          v2f bb = *(const v2f*)(xrow + kk);            // ds_load_b64
          acc = wmma_f32_4(a, bb, acc);
        }
        int mrow = mt * 16 + kh * 8;                    // C: vgpr r -> M=mrow+r
#pragma unroll
        for (int r = 0; r < 8; ++r)
          tbuf[(mrow + r) * XS + nb + m16] = acc[r];
      }
      __syncthreads();

      // ---- dw3x3 on center 16x16 + epilogue --------------------------------
      if (g < 2) {
        // thread -> (channel c, quarter sub); staggered p avoids LDS conflicts
        int c = tid >> 2, sub = tid & 3;
        int hd = c >> 3, cl = c & 7;
        int gc = g * 64 + c;
        float w9[9];
#pragma unroll
        for (int i = 0; i < 9; ++i) w9[i] = wdw[gc * 9 + i];
        float accN = 0.f;
        float accG[8] = {0.f, 0.f, 0.f, 0.f, 0.f, 0.f, 0.f, 0.f};
        for (int ii = 0; ii < 64; ++ii) {
          int p = (sub << 6) | ((ii + (sub << 4)) & 63);
          int cy = p >> 4, cx = p & 15;
          const float* tr = &tbuf[c * XS + cy * 18 + cx];
          float o = tr[0]  * w9[0] + tr[1]  * w9[1] + tr[2]  * w9[2]
                  + tr[18] * w9[3] + tr[19] * w9[4] + tr[20] * w9[5]
                  + tr[36] * w9[6] + tr[37] * w9[7] + tr[38] * w9[8];
          accN += o * o;
          if (s == 0) {
            (g == 0 ? qc : kc)[c * QS + p] = o;
          } else if (g == 0) {  // q2 : G2[ci=cl][cj] += q2*k1
#pragma unroll
            for (int j = 0; j < 8; ++j) accG[j] += o * kc[(hd * 8 + j) * QS + p];
          } else {              // k2 : G1[ci][cj=cl] += q1*k2
#pragma unroll
            for (int j = 0; j < 8; ++j) accG[j] += qc[(hd * 8 + j) * QS + p] * o;
          }
        }
        accN += __shfl_xor(accN, 1);
        accN += __shfl_xor(accN, 2);
        if (s == 1) {
#pragma unroll
          for (int j = 0; j < 8; ++j) {
            accG[j] += __shfl_xor(accG[j], 1);
            accG[j] += __shfl_xor(accG[j], 2);
          }
        }
        if (sub == 0) {
          atomicAdd(&norms[((s * 2 + g) * 4 + b) * 64 + c], accN);
          if (s == 1) {
            int dir = (g == 0) ? 1 : 0;  // q2.k1 -> attn2, q1.k2 -> attn1
            float* Gd = &G[((dir * 4 + b) * 8 + hd) * 64];
            if (g == 0) {
#pragma unroll
              for (int j = 0; j < 8; ++j) atomicAdd(&Gd[cl * 8 + j], accG[j]);
            } else {
#pragma unroll
              for (int j = 0; j < 8; ++j) atomicAdd(&Gd[j * 8 + cl], accG[j]);
            }
          }
        }
      } else {
        // v: per-channel iteration -> fully coalesced global stores
        for (int c = 0; c < 64; ++c) {
          int gc = 128 + c;
          float w9[9];
#pragma unroll
          for (int i = 0; i < 9; ++i) w9[i] = wdw[gc * 9 + i];
          int p = tid, cy = p >> 4, cx = p & 15;
          const float* tr = &tbuf[c * XS + cy * 18 + cx];
          float o = tr[0]  * w9[0] + tr[1]  * w9[1] + tr[2]  * w9[2]
                  + tr[18] * w9[3] + tr[19] * w9[4] + tr[20] * w9[5]
                  + tr[36] * w9[6] + tr[37] * w9[7] + tr[38] * w9[8];
          vout[((size_t)((s * 4 + b) * 64 + c) << 16) + ((ty + cy) << 8) + (tx + cx)] = o;
        }
      }
      __syncthreads();
    }
    __syncthreads();
  }
}

// ---------------------------------------------------------------------------
// Kernel 2: per (dir,b,head): softmax(G/(|q||k|)*temp) -> SVD (Jacobi on
// attn*attn^T) -> Orthmap(6) -> FRMap/Projmap sum -> mixing matrix M/4.
// ---------------------------------------------------------------------------
__global__ void k_gr(const float* __restrict__ G, const float* __restrict__ norms,
                     const float* __restrict__ temp,
                     const float* __restrict__ W1, const float* __restrict__ W2,
                     const float* __restrict__ W3, const float* __restrict__ W4,
                     float* __restrict__ Mout)  // [2][4][8][8][8]
{
  int u = threadIdx.x;
  if (u >= 64) return;
  int dir = u >> 5, idx = u & 31, b = idx >> 3, hd = idx & 7;
  int t4q = dir ? 2 : 0;  // dir0: nq1 ; dir1: nq2
  int t4k = dir ? 1 : 3;  // dir0: nk2 ; dir1: nk1
  const float* Gb = G + ((dir * 4 + b) * 8 + hd) * 64;

  float qn[8], kn[8];
  for (int i = 0; i < 8; ++i)
    qn[i] = fmaxf(sqrtf(norms[(t4q * 4 + b) * 64 + hd * 8 + i]), 1e-12f);
  for (int j = 0; j < 8; ++j)
    kn[j] = fmaxf(sqrtf(norms[(t4k * 4 + b) * 64 + hd * 8 + j]), 1e-12f);
  float tp = temp[hd];

  float attn[8][8];
  for (int i = 0; i < 8; ++i) {
    float row[8], mx = -1e30f;
    for (int j = 0; j < 8; ++j) {
      row[j] = Gb[i * 8 + j] / (qn[i] * kn[j]) * tp;
      mx = fmaxf(mx, row[j]);
    }
    float ss = 0.f;
    for (int j = 0; j < 8; ++j) { row[j] = expf(row[j] - mx); ss += row[j]; }
    float inv = 1.f / ss;
    for (int j = 0; j < 8; ++j) attn[i][j] = row[j] * inv;
  }

  // S = attn attn^T ; left singular vectors = eigvecs of S
  float S[8][8], V[8][8];
  for (int i = 0; i < 8; ++i)
    for (int j = 0; j < 8; ++j) {
      float acc = 0.f;
      for (int t = 0; t < 8; ++t) acc += attn[i][t] * attn[j][t];
      S[i][j] = acc;
      V[i][j] = (i == j) ? 1.f : 0.f;
    }
  // cyclic Jacobi
  for (int sweep = 0; sweep < 12; ++sweep)
    for (int p = 0; p < 7; ++p)
      for (int q = p + 1; q < 8; ++q) {
        float apq = S[p][q];
        if (fabsf(apq) <= 1e-13f) continue;
        float tau = (S[q][q] - S[p][p]) / (2.f * apq);
        float t = (tau >= 0.f) ? 1.f / (tau + sqrtf(1.f + tau * tau))
                               : -1.f / (-tau + sqrtf(1.f + tau * tau));
        float c_ = 1.f / sqrtf(1.f + t * t);
        float s_ = t * c_;
        for (int i = 0; i < 8; ++i) {
          float sip = S[i][p], siq = S[i][q];
          S[i][p] = c_ * sip - s_ * siq;
          S[i][q] = s_ * sip + c_ * siq;
        }
        for (int i = 0; i < 8; ++i) {
          float spi = S[p][i], sqi = S[q][i];
          S[p][i] = c_ * spi - s_ * sqi;
          S[q][i] = s_ * spi + c_ * sqi;
        }
        for (int i = 0; i < 8; ++i) {
          float vip = V[i][p], viq = V[i][q];
          V[i][p] = c_ * vip - s_ * viq;
          V[i][q] = s_ * vip + c_ * viq;
        }
      }
  // sort eigenpairs desc
  float d[8];
  for (int i = 0; i < 8; ++i) d[i] = S[i][i];
  for (int a = 0; a < 7; ++a) {
    int m = a;
    for (int i = a + 1; i < 8; ++i) if (d[i] > d[m]) m = i;
    if (m != a) {
      float td = d[a]; d[a] = d[m]; d[m] = td;
      for (int i = 0; i < 8; ++i) {
        float tv = V[i][a]; V[i][a] = V[i][m]; V[i][m] = tv;
      }
    }
  }
  // sign convention: largest-|component| positive
  for (int j = 0; j < 6; ++j) {
    int im = 0; float bm = fabsf(V[0][j]);
    for (int i = 1; i < 8; ++i) {
      float av = fabsf(V[i][j]);
      if (av > bm) { bm = av; im = i; }
    }
    if (V[im][j] < 0.f)
      for (int i = 0; i < 8; ++i) V[i][j] = -V[i][j];
  }

  // Grassmann: A += mask .* (Z Z^T), Z = (W U^T)^T ; M = A/4
  float A8[8][8];
  for (int i = 0; i < 8; ++i)
    for (int j = 0; j < 8; ++j) A8[i][j] = 0.f;
  const float* Ws[4] = {W1, W2, W3, W4};
  const int kr[4] = {5, 4, 3, 2};
  for (int w = 0; w < 4; ++w) {
    float Y[5][8];
    for (int t = 0; t < kr[w]; ++t)
      for (int n = 0; n < 8; ++n) {
        float acc = 0.f;
        for (int dd = 0; dd < 6; ++dd) acc += Ws[w][t * 6 + dd] * V[n][dd];
        Y[t][n] = acc;
      }
    for (int i = 0; i < 8; ++i)
      for (int j = 0; j < 8; ++j) {
        float zz = 0.f;
        for (int t = 0; t < kr[w]; ++t) zz += Y[t][i] * Y[t][j];
        A8[i][j] += (i == j) ? zz : -zz;
      }
  }
  float* Mo = Mout + ((dir * 4 + b) * 8 + hd) * 64;
  for (int i = 0; i < 8; ++i)
    for (int j = 0; j < 8; ++j) Mo[i * 8 + j] = A8[i][j] * 0.25f;
}

// ---------------------------------------------------------------------------
// Kernel 2b: fold proj conv into the mixing matrix:
//   E[oc,jc] = sum_ci wproj[oc, hd(jc)*8+ci] * M[hd(jc)][ci][jc%8]
// ---------------------------------------------------------------------------
__global__ __launch_bounds__(256) void k_E(const float* __restrict__ Mbuf,
                                           const float* __restrict__ wproj,
                                           float* __restrict__ E)  // [8][64][64]
{
  int sd = blockIdx.x;  // dir*4 + b
  for (int e = threadIdx.x; e < 4096; e += 256) {
    int oc = e >> 6, jc = e & 63, hd = jc >> 3, cl = jc & 7;
    const float* Mo = Mbuf + (sd * 8 + hd) * 64;
    float acc = 0.f;
#pragma unroll
    for (int ci = 0; ci < 8; ++ci)
      acc += wproj[oc * 64 + hd * 8 + ci] * Mo[ci * 8 + cl];
    E[sd * 4096 + e] = acc;
  }
}

// ---------------------------------------------------------------------------
// Kernel 3: out[64, HW] = E[64x64] @ v[64, HW] per (dir,b) via WMMA f32.
// Persistent over 4 chunks, TDM double-buffered: tile i+1 DMAs into the
// alternate LDS buffer while tile i is multiplied (s_wait_tensorcnt 1).
// ---------------------------------------------------------------------------
__global__ __launch_bounds__(256) void k_av(const float* __restrict__ v,
                                            const float* __restrict__ E,
                                            float* __restrict__ out)
{
  extern __shared__ float vt[];        // 2 buffers x 64 x VS
  const int tid = threadIdx.x;
  const int wg = blockIdx.x;           // 512 WGs: sd = wg>>6, 4 chunks each
  const int sd = wg >> 6;              // dir*4 + b
  const int c0 = (wg & 63) * 4;        // first of 4 chunks (256 px each)
  const float* vb = v + ((size_t)sd << 22);
  const float* Eb = E + sd * 4096;
  const unsigned ldsbase = (unsigned)(size_t)(void*)vt;
  const int wave = tid >> 5, lane = tid & 31, m16 = lane & 15, kh = lane >> 4;

  if (wave == 0) tdm_load_tile(vb + ((size_t)c0 << 8), ldsbase);
  for (int i = 0; i < 4; ++i) {
    int buf = i & 1;
    if (wave == 0) {
      if (i < 3) {
        tdm_load_tile(vb + ((size_t)(c0 + i + 1) << 8),
                      ldsbase + (unsigned)((buf ^ 1) * (64 * VS * 4)));
        __builtin_amdgcn_s_wait_tensorcnt(1);  // tile i done, i+1 in flight
      } else {
        __builtin_amdgcn_s_wait_tensorcnt(0);
      }
    }
    __syncthreads();

    const float* vtb = vt + buf * (64 * VS);
    size_t pxb = (size_t)(c0 + i) << 8;
    for (int t8 = wave; t8 < 64; t8 += 8) {  // wave-uniform: EXEC full
      int mt = t8 & 3, nt = t8 >> 2;
      const float* erow = &Eb[(mt * 16 + m16) * 64];
      v8f acc = {};
      for (int k0 = 0; k0 < 64; k0 += 4) {
        int kk = k0 + kh * 2;
        v2f a = *(const v2f*)(erow + kk);
        v2f bb;
        bb.x = vtb[kk * VS + nt * 16 + m16];
        bb.y = vtb[(kk + 1) * VS + nt * 16 + m16];
        acc = wmma_f32_4(a, bb, acc);
      }
      size_t ob = (size_t)sd << 22;
      int ocb = mt * 16 + kh * 8;
#pragma unroll
      for (int r = 0; r < 8; ++r)
        out[ob + ((size_t)(ocb + r) << 16) + pxb + nt * 16 + m16] = acc[r];
    }
    __syncthreads();
  }
}

// ---------------------------------------------------------------------------
extern "C" void kernel_launch(void* const* d_in, const int* in_sizes, int n_in,
                              void* d_out, int out_size, void* d_ws, size_t ws_size,
                              hipStream_t stream) {
  (void)in_sizes; (void)n_in; (void)out_size; (void)ws_size;
  const float* xir  = (const float*)d_in[0];
  const float* xvi  = (const float*)d_in[1];
  const float* wqkv = (const float*)d_in[2];
  const float* wdw  = (const float*)d_in[3];
  const float* wprj = (const float*)d_in[4];
  const float* temp = (const float*)d_in[5];
  const float* W1   = (const float*)d_in[6];
  const float* W2   = (const float*)d_in[7];
  const float* W3   = (const float*)d_in[8];
  const float* W4   = (const float*)d_in[9];
  float* out = (float*)d_out;

  float* v     = (float*)d_ws;        // 2*4*64*65536 floats
  float* G     = v + 33554432;        // 4096 floats
  float* norms = G + 4096;            // 1024 floats
  float* Mbuf  = norms + 1024;        // 4096 floats
  float* E     = Mbuf + 4096;         // 32768 floats

  (void)hipMemsetAsync(G, 0, (4096 + 1024) * sizeof(float), stream);

  size_t lds1 = (size_t)(336 * PS + 64 * XS + 2 * 64 * QS) * sizeof(float);  // ~310 KB
  k_qkv<<<dim3(1024), dim3(256), lds1, stream>>>(xir, xvi, wqkv, wdw, v, G, norms);
  k_gr<<<dim3(1), dim3(64), 0, stream>>>(G, norms, temp, W1, W2, W3, W4, Mbuf);
  k_E<<<dim3(8), dim3(256), 0, stream>>>(Mbuf, wprj, E);
  size_t lds3 = (size_t)(2 * 64 * VS) * sizeof(float);  // 136 KB (double buffer)
  k_av<<<dim3(512), dim3(256), lds3, stream>>>(v, E, out);
}